// embedding_layer_57346403336316
// MI455X (gfx1250) — compile-verified
//
#include <hip/hip_runtime.h>
#include <stdint.h>
#include <stddef.h>

// ---------------------------------------------------------------------------
// max-norm-clipped embedding update, D = 128, f32.
//   out[t,:] = W[t,:] * (||W[t]|| > 1 ? 1/(||W[t]||+1e-7) : 1) + (mask[t] ? x[g[t],:] : 0)
// Memory-bound: 281.6 MB total traffic -> ~12.1 us floor @ 23.3 TB/s.
// CDNA5 paths used: global_load_async_to_lds_b128 (ASYNCcnt) for W tiles,
// v_wmma_f32_16x16x4_f32 Gram trick for the 16 row-norms per wave, NT hints.
// ---------------------------------------------------------------------------

typedef float v2f __attribute__((ext_vector_type(2)));
typedef float v4f __attribute__((ext_vector_type(4)));
typedef float v8f __attribute__((ext_vector_type(8)));

#define D_FIX        128
#define ROW_DW_PAD   132                 // 128 dwords + 4 dword pad (bank-conflict-free)
#define ROW_B_PAD    (ROW_DW_PAD * 4)    // 528 bytes
#define WAVES_PER_BLOCK 4
#define TILE_ROWS    16

// ---------------- setup: rank[t] = j+1 where in_idx[j] == t (0 = unmasked) --
__global__ void en_zero_rank_kernel(int* __restrict__ rank, int T) {
    int i = blockIdx.x * blockDim.x + threadIdx.x;
    if (i < T) rank[i] = 0;
}

__global__ void en_scatter_rank_kernel(const int* __restrict__ in_idx,
                                       int* __restrict__ rank, int n_in, int T) {
    int j = blockIdx.x * blockDim.x + threadIdx.x;
    if (j < n_in) {
        int t = in_idx[j];
        if (t >= 0 && t < T) rank[t] = j + 1;
    }
}

// ---------------- main streaming kernel: one wave = 16 rows x 128 cols ------
__launch_bounds__(WAVES_PER_BLOCK * 32)
__global__ void en_maxnorm_embed_kernel(const float* __restrict__ W,
                                        const float* __restrict__ x,
                                        const int* __restrict__ rank,
                                        float* __restrict__ out,
                                        int T, int n_in, int use_rank) {
    __shared__ __align__(16) float smem[WAVES_PER_BLOCK * TILE_ROWS * ROW_DW_PAD];

    const int lane        = threadIdx.x & 31;
    const int waveInBlock = threadIdx.x >> 5;
    const int tile        = blockIdx.x * WAVES_PER_BLOCK + waveInBlock;
    const int tileStart   = tile * TILE_ROWS;
    if (tileStart >= T) return;

    float* wtile = smem + waveInBlock * (TILE_ROWS * ROW_DW_PAD);

    if (tileStart + TILE_ROWS <= T) {
        // ---- async copy: 16 contiguous rows (8 KB) of W -> padded LDS tile.
        // Per instruction the wave moves 512 contiguous bytes; 16 instructions
        // cover the tile; ASYNCcnt tracks completion.
        const uint32_t ldsBase = (uint32_t)(uintptr_t)wtile;   // low 32b of flat ptr = LDS offset
        const uint64_t gBase   = (uint64_t)(uintptr_t)(W + (size_t)tileStart * D_FIX);
#pragma unroll
        for (int i = 0; i < TILE_ROWS; ++i) {
            uint32_t ldsAddr = ldsBase + (uint32_t)(i * ROW_B_PAD + lane * 16);
            uint64_t gAddr   = gBase + (uint64_t)(i * 512 + lane * 16);
            asm volatile("global_load_async_to_lds_b128 %0, %1, off"
                         :: "v"(ldsAddr), "v"(gAddr) : "memory");
        }
#if __has_builtin(__builtin_amdgcn_s_wait_asynccnt)
        __builtin_amdgcn_s_wait_asynccnt(0);
#else
        asm volatile("s_wait_asynccnt 0x0" ::: "memory");
#endif

#if __has_builtin(__builtin_amdgcn_wmma_f32_16x16x4_f32)
        // ---- row norms via Gram trick: acc = A * A^T over K=128 in 32 chunks.
        // f32 16x16x4 A layout: lane m (0-15): {K0,K1}; lane m+16: {K2,K3} —
        // identical to the B=A^T layout, so the same regs feed both operands.
        v8f acc = {};
        const int m = lane & 15;
        const int e = (lane < 16) ? 0 : 2;     // dword sub-offset within K-chunk
#pragma unroll
        for (int c = 0; c < 32; ++c) {
            const v2f a = *(const v2f*)(wtile + m * ROW_DW_PAD + c * 4 + e);
            acc = __builtin_amdgcn_wmma_f32_16x16x4_f32(
                false, a, false, a, (short)0, acc, false, false);
        }
#endif

        // ---- scale, add x where masked, store (all 128b, fully coalesced) --
#pragma unroll
        for (int r = 0; r < TILE_ROWS; ++r) {
            const v4f w = *(const v4f*)(wtile + r * ROW_DW_PAD + lane * 4);
#if __has_builtin(__builtin_amdgcn_wmma_f32_16x16x4_f32)
            // diag(r,r) of D-layout: r<8 -> VGPR r, lane r ; r>=8 -> VGPR r-8, lane r+16
            float norm2 = (r < 8) ? __shfl(acc[r], r, 32)
                                  : __shfl(acc[r - 8], r + 16, 32);
#else
            float s = w.x * w.x + w.y * w.y + w.z * w.z + w.w * w.w;
#pragma unroll
            for (int d = 16; d > 0; d >>= 1) s += __shfl_xor(s, d, 32);
            float norm2 = s;
#endif
            float scale = 1.0f;
            if (norm2 > 1.0f) scale = 1.0f / (sqrtf(norm2) + 1e-7f);

            const int row = tileStart + r;
            int g; bool msk;
            if (use_rank) { int rk = rank[row]; msk = (rk != 0); g = rk - 1; }
            else          { msk = (row < n_in); g = row; }

            v4f o = w * scale;
            if (msk) {
                const v4f xv = __builtin_nontemporal_load(
                    (const v4f*)(x + (size_t)g * D_FIX) + lane);
                o += xv;
            }
            __builtin_nontemporal_store(o, (v4f*)(out + (size_t)row * D_FIX) + lane);
        }
    } else {
        // ---- partial tail tile: plain wave-cooperative per-row path --------
        for (int r = 0; r < TILE_ROWS && tileStart + r < T; ++r) {
            const int row = tileStart + r;
            const v4f w = *(const v4f*)(W + (size_t)row * D_FIX + lane * 4);
            float s = w.x * w.x + w.y * w.y + w.z * w.z + w.w * w.w;
#pragma unroll
            for (int d = 16; d > 0; d >>= 1) s += __shfl_xor(s, d, 32);
            float scale = (s > 1.0f) ? 1.0f / (sqrtf(s) + 1e-7f) : 1.0f;

            int g; bool msk;
            if (use_rank) { int rk = rank[row]; msk = (rk != 0); g = rk - 1; }
            else          { msk = (row < n_in); g = row; }

            v4f o = w * scale;
            if (msk) o += *(const v4f*)(x + (size_t)g * D_FIX + lane * 4);
            *(v4f*)(out + (size_t)row * D_FIX + lane * 4) = o;
        }
    }
}

// ---------------- generic-D safety net (one block per row) ------------------
__global__ void en_maxnorm_generic_kernel(const float* __restrict__ W,
                                          const float* __restrict__ x,
                                          const int* __restrict__ rank,
                                          float* __restrict__ out,
                                          int T, int n_in, int D, int use_rank) {
    __shared__ float red[256];
    for (int row = blockIdx.x; row < T; row += gridDim.x) {
        float s = 0.0f;
        for (int d = threadIdx.x; d < D; d += blockDim.x) {
            float v = W[(size_t)row * D + d];
            s += v * v;
        }
        red[threadIdx.x] = s;
        __syncthreads();
        for (int k = blockDim.x / 2; k > 0; k >>= 1) {
            if ((int)threadIdx.x < k) red[threadIdx.x] += red[threadIdx.x + k];
            __syncthreads();
        }
        float n2 = red[0];
        __syncthreads();
        float scale = (n2 > 1.0f) ? 1.0f / (sqrtf(n2) + 1e-7f) : 1.0f;

        int g; bool msk;
        if (use_rank) { int rk = rank[row]; msk = (rk != 0); g = rk - 1; }
        else          { msk = (row < n_in); g = row; }

        for (int d = threadIdx.x; d < D; d += blockDim.x) {
            float o = W[(size_t)row * D + d] * scale;
            if (msk) o += x[(size_t)g * D + d];
            out[(size_t)row * D + d] = o;
        }
        __syncthreads();
    }
}

// ---------------------------------------------------------------------------
extern "C" void kernel_launch(void* const* d_in, const int* in_sizes, int n_in_args,
                              void* d_out, int out_size, void* d_ws, size_t ws_size,
                              hipStream_t stream) {
    const int*   in_idx = (const int*)d_in[0];
    const int    n_in   = in_sizes[0];
    const int    n_off  = in_sizes[1];
    const float* x      = (const float*)d_in[2];
    const float* W      = (const float*)d_in[3];
    float*       out    = (float*)d_out;

    const int T = n_in + n_off;
    if (T <= 0) return;
    const int D = in_sizes[3] / T;

    int* rank = (int*)d_ws;
    const int use_rank = (d_ws != nullptr && ws_size >= (size_t)T * sizeof(int)) ? 1 : 0;
    if (use_rank) {
        en_zero_rank_kernel<<<(T + 255) / 256, 256, 0, stream>>>(rank, T);
        en_scatter_rank_kernel<<<(n_in + 255) / 256, 256, 0, stream>>>(in_idx, rank, n_in, T);
    }

    if (D == D_FIX) {
        const int tiles  = (T + TILE_ROWS - 1) / TILE_ROWS;
        const int blocks = (tiles + WAVES_PER_BLOCK - 1) / WAVES_PER_BLOCK;
        en_maxnorm_embed_kernel<<<blocks, WAVES_PER_BLOCK * 32, 0, stream>>>(
            W, x, rank, out, T, n_in, use_rank);
    } else {
        int blocks = (T < 4096) ? T : 4096;
        en_maxnorm_generic_kernel<<<blocks, 256, 0, stream>>>(
            W, x, rank, out, T, n_in, D, use_rank);
    }
}